// NeuroNN_14465449853182
// MI455X (gfx1250) — compile-verified
//
#include <hip/hip_runtime.h>

// ---------------------------------------------------------------------------
// NeuroNN fixed-point solver for MI455X (gfx1250, wave32, WMMA).
//   Per step: [96x2000] @ [2000x2000]^T for W and W*W, fused with the
//   Ricciardi transfer + Euler update + convergence metric. 60 steps.
//   f32 WMMA 16x16x4. W (16MB) stays resident in the 192MB L2; the W^2
//   B-fragment is squared in-register. b128 loads feed 2 WMMAs each via a
//   K-permutation (K-sets {k,k+1,k+4,k+5} / {k+2,k+3,k+6,k+7}).
// ---------------------------------------------------------------------------

typedef __attribute__((ext_vector_type(2))) float v2f;
typedef __attribute__((ext_vector_type(4))) float v4f;
typedef __attribute__((ext_vector_type(8))) float v8f;

#define NN   2000
#define NE   1600
#define MM   96
#define KDEG 0.017453292519943295f   // pi/180

// ---------------- Ricciardi transfer function --------------------------------
__device__ __forceinline__ float f_ricci(float x) {
  const float t = -x / (1.0f + x);
  float p = 0.14805913578876898f;
  p = p * t + 0.64290613877355551f;
  p = p * t + 1.0616084849547165f;
  p = p * t + 0.93524391761244940f;
  p = p * t + 0.62718906618071668f;
  p = p * t + 0.32171431660633076f;
  p = p * t + 0.32056016125642045f;
  p = p * t + 0.77373949685442023f;
  p = p * t + 0.22757881388024176f;
  return __logf(2.0f * x + 1.0f) + p * t;
}

__device__ __forceinline__ float g_ricci(float x) {
  const float z = x / (2.0f + x);
  float en = -155.73340457809226f;
  en = en * z + 456.58245777026514f;
  en = en * z - 520.37554793489681f;
  en = en * z + 279.56761105465944f;
  en = en * z - 56.532378057580381f;
  en = en * z - 7.0529131065835378f;
  en = en * z + 3.5441754117462949f;
  en = en * z;
  float de = -16.445022798669722f;
  de = de * z + 277.18420330693891f;
  de = de * z - 599.58577549598340f;
  de = de * z + 601.08633903294185f;
  de = de * z - 334.20436223415163f;
  de = de * z + 98.656602235468327f;
  de = de * z - 7.2984226138266743f;
  de = de * z - 4.1357968834226053f;
  de = de * z + 1.0f;
  return __fdividef(en, de);
}

__device__ __forceinline__ float phi_fn(float mu, float sig, float tau_ref) {
  const float rs = __frcp_rn(sig);
  const float xp = mu * rs;
  const float xm = (mu - 20.0f) * rs;   // V_THR = 20
  float r;
  if (xm > 0.0f) {
    r = __fdividef(1.0f, f_ricci(xp) - f_ricci(xm));
  } else if (xp > 0.0f) {
    const float nxm = fminf(-xm, 9.0f);
    r = __fdividef(1.0f, f_ricci(xp) + __expf(nxm * nxm) * g_ricci(nxm));
  } else {
    const float nxp = fminf(fmaxf(-xp, 0.0f), 9.0f);
    const float nxm = fminf(-xm, 9.0f);
    const float arg = g_ricci(nxm) - __expf(nxp * nxp - nxm * nxm) * g_ricci(nxp);
    r = __expf(-nxm * nxm - __logf(fmaxf(arg, 1e-12f)));
  }
  // rate = 1 / (tau_ref + tau / r),  tau = 0.01 for E and I
  return __fdividef(1.0f, tau_ref + 0.01f * __fdividef(1.0f, fmaxf(r, 1e-12f)));
}

// ---------------- weight / input construction --------------------------------
__device__ __forceinline__ float pref_of(int idx) {
  return (idx < NE) ? (float)idx * (179.99f / 1599.0f)
                    : (float)(idx - NE) * (179.99f / 399.0f);
}

__global__ void build_weights(const float* __restrict__ jp,
                              const float* __restrict__ pp,
                              const float* __restrict__ wp,
                              const float* __restrict__ rnd,
                              float* __restrict__ W) {
  const int j = blockIdx.x * blockDim.x + threadIdx.x;
  const int i = blockIdx.y;
  if (j >= NN) return;
  const int conn = 2 * (j >= NE) + (i >= NE);      // 0:ee 1:ei 2:ie 3:ii
  const float J  = __expf(jp[conn]);
  const float P  = __fdividef(1.0f, 1.0f + __expf(-2.0f * __expf(pp[conn])));
  const float Wd = __expf(wp[conn]);
  const float diff  = fabsf(pref_of(i) - pref_of(j));
  const float denom = 4.0f * (KDEG * Wd) * (KDEG * Wd);
  const float z = __expf(__fdividef(__cosf(2.0f * KDEG * diff) - 1.0f, denom));
  const float rv = __builtin_nontemporal_load(rnd + i * NN + j);  // one-shot read
  const float s = __fdividef(1.0f, 1.0f + __expf(-32.0f * (P * z - rv)));
  const float sgn = (j < NE) ? 1.0f : -1.0f;       // Dale's law by presynaptic col
  W[i * NN + j] = sgn * J * s;
}

__global__ void build_inmean(float* __restrict__ inmean) {
  const int n = blockIdx.x * blockDim.x + threadIdx.x;
  const int c = blockIdx.y;                        // 0..95, c = contrast*12 + ori
  if (n >= NN) return;
  const float CONTR[8] = {0.0f, 0.0432773f, 0.103411f, 0.186966f,
                          0.303066f, 0.464386f, 0.68854f, 1.0f};
  const float ori = 15.0f * (float)(c % 12);
  const float denom = 4.0f * (KDEG * 30.0f) * (KDEG * 30.0f);  // w_ff = 30
  const float g = __expf(__fdividef(
      __cosf(2.0f * KDEG * (ori - pref_of(n))) - 1.0f, denom));
  inmean[c * NN + n] = CONTR[c / 12] * 20.0f * g;
}

__global__ void zero_kernel(float* __restrict__ p, int n) {
  const int idx = blockIdx.x * blockDim.x + threadIdx.x;
  if (idx < n) p[idx] = 0.0f;
}

// ---------------- fused step: 2x WMMA GEMM + Phi + metric --------------------
// grid = 750 tiles (6 cond-tiles x 125 neuron-tiles), block = 1 wave (32 lanes)
__global__ __launch_bounds__(32) void step_kernel(
    const float* __restrict__ W,
    const float* __restrict__ inmean,
    const float* __restrict__ r_cur, float* __restrict__ r_next,
    unsigned int* __restrict__ m_step /* 96 uint-bits-of-float, pre-zeroed */) {
  const int lane = threadIdx.x;
  const int tM = blockIdx.x % 6;           // condition tile
  const int tN = blockIdx.x / 6;           // neuron tile
  const int c0 = tM * 16;
  const int i0 = tN * 16;
  const int hf  = lane >> 4;               // 0 | 1
  const int l15 = lane & 15;

  // Per 8 K-values each lane loads one float4 per matrix:
  //   half 0 at row+k   -> .xy = K-slots (k,k+1)   [inst1], .zw = (k+2,k+3) [inst2]
  //   half 1 at row+k+4 -> .xy = K-slots (k+4,k+5) [inst1], .zw = (k+6,k+7) [inst2]
  // inst1 contracts K-set {k,k+1,k+4,k+5}; inst2 {k+2,k+3,k+6,k+7}; A and B use
  // the same permutation so the sum over K is unchanged.
  const float* aptr = r_cur + (c0 + l15) * NN + 4 * hf;   // A: r[M=cond][K=j]
  const float* bptr = W     + (i0 + l15) * NN + 4 * hf;   // B: W^T, B[k=j][n=i]

  v8f accW = {};   // r @ W.T     tile
  v8f accV = {};   // r @ (W*W).T tile -- B fragment squared in-register
#pragma unroll 8
  for (int k = 0; k < NN; k += 8) {
    const v4f a  = *(const v4f*)(aptr + k);
    const v4f b  = *(const v4f*)(bptr + k);
    const v4f b2 = b * b;                  // elementwise (W^2) fragment
    const v2f a_lo  = __builtin_shufflevector(a,  a,  0, 1);
    const v2f a_hi  = __builtin_shufflevector(a,  a,  2, 3);
    const v2f b_lo  = __builtin_shufflevector(b,  b,  0, 1);
    const v2f b_hi  = __builtin_shufflevector(b,  b,  2, 3);
    const v2f b2_lo = __builtin_shufflevector(b2, b2, 0, 1);
    const v2f b2_hi = __builtin_shufflevector(b2, b2, 2, 3);
    accW = __builtin_amdgcn_wmma_f32_16x16x4_f32(false, a_lo, false, b_lo,
                                                 (short)0, accW, false, false);
    accV = __builtin_amdgcn_wmma_f32_16x16x4_f32(false, a_lo, false, b2_lo,
                                                 (short)0, accV, false, false);
    accW = __builtin_amdgcn_wmma_f32_16x16x4_f32(false, a_hi, false, b_hi,
                                                 (short)0, accW, false, false);
    accV = __builtin_amdgcn_wmma_f32_16x16x4_f32(false, a_hi, false, b2_hi,
                                                 (short)0, accV, false, false);
  }

  // Epilogue. C layout: n = l15, m = v + 8*hf.
  const int i = i0 + l15;
  const float tau_ref = (i < NE) ? 0.005f : 0.001f;   // TAU_REF: E=0.005, I=0.001
  const float dt_tinv = (i < NE) ? 0.1f   : 0.2f;     // DT * T_INV

  float metric[8];
#pragma unroll
  for (int v = 0; v < 8; ++v) {
    const int c = c0 + v + 8 * hf;
    const float mu  = 0.01f * accW[v] + inmean[c * NN + i];   // TAU = 0.01
    const float var = 0.01f * accV[v] + 25.0f;                // sig_ext^2 = 25
    const float sig = __builtin_amdgcn_sqrtf(var);            // var >= 25
    const float ph  = phi_fn(mu, sig, tau_ref);
    const float ro  = r_cur[c * NN + i];
    const float dr  = dt_tinv * (ph - ro);
    const float rn  = ro + dr;
    r_next[c * NN + i] = rn;
    metric[v] = fabsf(dr) * __frcp_rn(fmaxf(fabsf(rn), 1.0f)) * 1e5f; // /XTOL
  }
  // max over the 16 lanes of each half (they share the same 8 condition rows)
#pragma unroll
  for (int off = 1; off < 16; off <<= 1) {
#pragma unroll
    for (int v = 0; v < 8; ++v)
      metric[v] = fmaxf(metric[v], __shfl_xor(metric[v], off, 32));
  }
  if (l15 == 0) {
#pragma unroll
    for (int v = 0; v < 8; ++v)
      atomicMax(&m_step[c0 + v + 8 * hf], __float_as_uint(metric[v]));
  }
}

// ---------------- finalize: transpose + convergence average ------------------
__global__ void transpose_out(const float* __restrict__ r, float* __restrict__ out) {
  const int idx = blockIdx.x * blockDim.x + threadIdx.x;
  if (idx >= MM * NN) return;
  const int n = idx / MM;
  const int c = idx % MM;
  out[idx] = r[c * NN + n];   // tuning[n, contrast, ori] = r_fp[c, n]
}

__global__ void avg_kernel(const float* __restrict__ marr, float* __restrict__ out) {
  __shared__ float sd[256];
  const int t = threadIdx.x;
  float s = 0.0f;
  for (int idx = 45 * MM + t; idx < 60 * MM; idx += 256) s += marr[idx];
  sd[t] = s;
  __syncthreads();
  for (int st = 128; st > 0; st >>= 1) {
    if (t < st) sd[t] += sd[t + st];
    __syncthreads();
  }
  if (t == 0) out[0] = sd[0] / (15.0f * (float)MM);
}

// ---------------------------------------------------------------------------
extern "C" void kernel_launch(void* const* d_in, const int* in_sizes, int n_in,
                              void* d_out, int out_size, void* d_ws, size_t ws_size,
                              hipStream_t stream) {
  const float* jp  = (const float*)d_in[0];
  const float* pp  = (const float*)d_in[1];
  const float* wp  = (const float*)d_in[2];
  const float* rnd = (const float*)d_in[3];

  float* ws   = (float*)d_ws;
  float* W    = ws;                 // 4,000,000 f32 (16 MB), L2-resident
  float* inm  = ws + 4000000;       //   192,000 f32
  float* rA   = ws + 4192000;       //   192,000 f32
  float* rB   = ws + 4384000;       //   192,000 f32
  float* marr = ws + 4576000;       //     5,760 f32 (60 steps x 96 conds)

  build_weights<<<dim3(8, NN), 256, 0, stream>>>(jp, pp, wp, rnd, W);
  build_inmean <<<dim3(8, MM), 256, 0, stream>>>(inm);
  // zero rA, rB, marr (contiguous: 389,760 floats)
  zero_kernel<<<(389760 + 255) / 256, 256, 0, stream>>>(rA, 389760);

  float* bufs[2] = {rA, rB};
  for (int s = 0; s < 60; ++s) {
    step_kernel<<<750, 32, 0, stream>>>(
        W, inm, bufs[s & 1], bufs[(s + 1) & 1],
        (unsigned int*)(marr + s * MM));
  }
  // 60 steps: final rates land in bufs[0]
  transpose_out<<<(MM * NN + 255) / 256, 256, 0, stream>>>(bufs[0], (float*)d_out);
  avg_kernel<<<1, 256, 0, stream>>>(marr, (float*)d_out + MM * NN);
}